// WaterNetSingle_16690242913227
// MI455X (gfx1250) — compile-verified
//
#include <hip/hip_runtime.h>
#include <hip/hip_bf16.h>
#include <cstddef>

#define NT 1000
#define NS 4096
#define NH 64
#define PI_F 3.1415f

typedef __attribute__((ext_vector_type(2))) float v2f;
typedef __attribute__((ext_vector_type(8))) float v8f;

// One block = 128 threads = 4 waves; block owns 16 cells, wave w owns h in [16w, 16w+16).
// Lane l: cell = blockBase + (l & 15); within each k-chunk of 4 h, lanes 0-15 take K=0,1
// and lanes 16-31 take K=2,3 (matches f32 WMMA A/B VGPR layouts).
__global__ __launch_bounds__(128, 1) void waternet_scan_kernel(
    const float* __restrict__ xg, const float* __restrict__ w, float* __restrict__ out)
{
    const int tid  = threadIdx.x;
    const int wid  = tid >> 5;        // 0..3 -> h base
    const int lane = tid & 31;
    const int half = lane >> 4;       // 0: K=0,1  1: K=2,3
    const int col  = lane & 15;       // WMMA column = cell within block
    const int cellBase = blockIdx.x * 16;
    const int cell  = cellBase + col;
    const int hbase = wid * 16;

    // ---------------- one-time weight preprocessing (all in registers) ----------------
    float sumExp = 0.0f;
    for (int h = 0; h < NH; ++h) sumExp += expf(w[6 * NH + h]);
    const float invSum = 1.0f / sumExp;

    float gm[8], ge[8], k2[8], k23[8], k3[8], gl[8], ga[8], qb[8], vi[8];
#pragma unroll
    for (int j = 0; j < 4; ++j) {
#pragma unroll
        for (int s = 0; s < 2; ++s) {
            const int i = 2 * j + s;
            const int h = hbase + 4 * j + 2 * half + s;
            gm[i]  = expf(w[0 * NH + h]) + 1.0f;
            ge[i]  = 2.0f / (1.0f + expf(-w[1 * NH + h]));
            k2[i]  = 1.0f / (1.0f + expf(-w[2 * NH + h]));
            k23[i] = 1.0f / (1.0f + expf(-w[3 * NH + h]));
            k3[i]  = 0.1f / (1.0f + expf(-w[4 * NH + h]));
            gl[i]  = 2.0f * expf(w[5 * NH + h]);
            ga[i]  = expf(w[6 * NH + h]) * invSum;
            qb[i]  = fmaxf(w[7 * NH + h], 0.0f);
            vi[i]  = fmaxf(w[8 * NH + h], 0.0f);
        }
    }

    // ---------------- state (register resident) ----------------
    float S0[8], S2[8], S3[8];
#pragma unroll
    for (int i = 0; i < 8; ++i) { S0[i] = 0.0f; S2[i] = 0.0f; S3[i] = 0.0f; }

    __shared__ float partY[2][4][16];   // double-buffered cross-wave partials

    const float4* __restrict__ xin = (const float4*)xg;  // [NT][NS] float4
    float4 xnext = xin[(size_t)0 * NS + cell];           // pipelined load for t=0

    for (int t = 0; t < NT; ++t) {
        const float4 xc = xnext;
        const int tn = (t + 1 < NT) ? (t + 1) : (NT - 1);
        xnext = xin[(size_t)tn * NS + cell];
        const int tp = (t + 8 < NT) ? (t + 8) : (NT - 1);
        __builtin_prefetch(&xin[(size_t)tp * NS + cell], 0, 1);  // global_prefetch_b8

        // per-cell rain/snow split (identical algebra to reference)
        const float P = xc.x, E = xc.y, T1 = xc.z, T2 = xc.w;
        const float Ta   = 0.5f * (T1 + T2);
        const bool  mid  = (T1 < 0.0f) && (T2 > 0.0f);
        const float arg  = mid ? (T1 + T2) / (T2 - T1) : 0.0f;
        float rP = 1.0f - acosf(arg) * (1.0f / PI_F);
        rP = (T1 >= 0.0f) ? 1.0f : ((T2 <= 0.0f) ? 0.0f : rP);
        const float Ps  = (1.0f - rP) * P;
        const float Pl  = rP * P;
        const float TaP = fmaxf(Ta, 0.0f);   // gm > 0 so relu(Ta*gm) = gm * relu(Ta)

        v8f acc = {};  // D accumulator: Y partial for this wave's 16 h
#pragma unroll
        for (int j = 0; j < 4; ++j) {
            v2f bz, av;
#pragma unroll
            for (int s = 0; s < 2; ++s) {
                const int i = 2 * j + s;
                const float H0  = S0[i] + Ps;
                const float qSm = fminf(H0, gm[i] * TaP);
                const float qIn = qSm + Pl * vi[i] - E * ge[i];
                const float t2v = S2[i] + qIn;
                const float Q1  = fmaxf(t2v - gl[i], 0.0f);
                const float H2  = fminf(fmaxf(t2v, 0.0f), gl[i]);
                const float q2  = H2 * k2[i];
                const float Q2  = q2 * (1.0f - k23[i]);
                const float H3  = S3[i] + q2 * k23[i];
                const float Q3  = fmaxf(H3 * k3[i] + qb[i], 0.0f);
                S0[i] = H0 - qSm;
                S2[i] = H2 - Q2;
                S3[i] = H3 - Q3;
                bz[s] = Q1 + Q2 + Q3;   // B[k][n] = Z[cell n][h_k]
                av[s] = ga[i];          // A[m][k] = ga[h_k] (row-replicated)
            }
            // D[m][n] += sum_k ga[h_k] * Z[cell n][h_k]   (fp32 WMMA, full precision)
            acc = __builtin_amdgcn_wmma_f32_16x16x4_f32(
                false, av, false, bz, (short)0, acc, false, false);
        }

        // every lane of the wave holds the same partial Y[cell col] in acc[0]
        partY[t & 1][wid][col] = acc[0];
        __syncthreads();
        if (tid < 16) {
            const float y = partY[t & 1][0][tid] + partY[t & 1][1][tid]
                          + partY[t & 1][2][tid] + partY[t & 1][3][tid];
            out[(size_t)t * NS + cellBase + tid] = y;
        }
        // no second barrier needed: next iteration writes the other LDS buffer
    }
}

extern "C" void kernel_launch(void* const* d_in, const int* in_sizes, int n_in,
                              void* d_out, int out_size, void* d_ws, size_t ws_size,
                              hipStream_t stream) {
    const float* x = (const float*)d_in[0];   // [NT, NS, 4] f32
    const float* w = (const float*)d_in[1];   // [576] f32
    float* out = (float*)d_out;               // [NT, NS] f32
    (void)in_sizes; (void)n_in; (void)out_size; (void)d_ws; (void)ws_size;
    dim3 grid(NS / 16);   // 256 blocks
    dim3 block(128);      // 4 waves
    hipLaunchKernelGGL(waternet_scan_kernel, grid, block, 0, stream, x, w, out);
}